// CGIterator_65000035058297
// MI455X (gfx1250) — compile-verified
//
#include <hip/hip_runtime.h>

typedef float v2f __attribute__((ext_vector_type(2)));
typedef float v8f __attribute__((ext_vector_type(8)));

#define NN        50000
#define KCH       128      // K
#define PCH       256      // 2K
#define NITER     4
#define NPB       16       // nodes per block
#define NTHREADS  256
#define NWAVES    8
#define EPSV      1e-6f

// Symmetric pair decode tables for the CG quadratic form: q -> (a,b), a<=b, 45 pairs (padded to 48).
constexpr int PA[48] = {0,0,0,0,0,0,0,0,0, 1,1,1,1,1,1,1,1, 2,2,2,2,2,2,2,
                        3,3,3,3,3,3, 4,4,4,4,4, 5,5,5,5, 6,6,6, 7,7, 8, 0,0,0};
constexpr int PB[48] = {0,1,2,3,4,5,6,7,8, 1,2,3,4,5,6,7,8, 2,3,4,5,6,7,8,
                        3,4,5,6,7,8, 4,5,6,7,8, 5,6,7,8, 6,7,8, 7,8, 8, 0,0,0};

__device__ __forceinline__ int l_of_m(int m) { return (m == 0) ? 0 : ((m < 4) ? 1 : 2); }

__global__ __launch_bounds__(NTHREADS, 1)
void cg_fused_kernel(const float* __restrict__ f0,
                     const float* __restrict__ f1,
                     const float* __restrict__ f2,
                     const float* __restrict__ Ug,
                     const float* __restrict__ gamma,
                     const float* __restrict__ Win,
                     const float* __restrict__ Wout,
                     float* __restrict__ out)
{
  extern __shared__ float lds[];
  float* feat = lds;                       // [NPB][9][KCH]  (node*9+m)*K + k   (residual)
  float* an   = feat + NPB * 9 * KCH;      // [9][NPB][KCH]  (m*NPB+node)*K + k (normalized)
  float* hc   = an   + 9 * NPB * KCH;      // [9][NPB][PCH]  (m*NPB+node)*P + p (2K intermediate)
  float* Ub   = hc   + 9 * NPB * PCH;      // [48][16] symmetrized CG as B-matrix (K=48, N=16)

  const int tid    = threadIdx.x;
  const int lane   = tid & 31;
  const int wave   = tid >> 5;
  const int node0  = blockIdx.x * NPB;

  // WMMA f32 16x16x4 fragment lane decomposition:
  //   A: lanes 0-15 hold K={0,1}, lanes 16-31 hold K={2,3}; row M = lane&15
  //   B mirrored; C/D: VGPR j -> M = j + 8*(lane>>4), N = lane&15
  const int row16  = lane & 15;
  const int khalf  = (lane >> 4) << 1;   // 0 or 2
  const int mibase = (lane >> 4) << 3;   // 0 or 8
  const bool hiHalf = (khalf != 0);

  // ---- Build symmetrized CG B-matrix once: Ub[kk][c] (zero padded kk>=45, c>=9)
  for (int idx = tid; idx < 48 * 16; idx += NTHREADS) {
    int kk = idx >> 4, c = idx & 15;
    float val = 0.f;
    if (kk < 45 && c < 9) {
      int a = PA[kk], b = PB[kk];
      val = (a == b) ? Ug[(a * 9 + a) * 9 + c]
                     : (Ug[(a * 9 + b) * 9 + c] + Ug[(b * 9 + a) * 9 + c]);
    }
    Ub[idx] = val;
  }

  // ---- Load features once (packed m order: l=0 -> m=0, l=1 -> m=1..3, l=2 -> m=4..8)
  for (int idx = tid; idx < NPB * KCH; idx += NTHREADS) {
    int node = idx / KCH, k = idx % KCH;
    feat[(node * 9 + 0) * KCH + k] = f0[(size_t)(node0 + node) * KCH + k];
  }
  for (int idx = tid; idx < NPB * 3 * KCH; idx += NTHREADS) {
    int k = idx % KCH; int m = (idx / KCH) % 3; int node = idx / (3 * KCH);
    feat[(node * 9 + 1 + m) * KCH + k] = f1[((size_t)(node0 + node) * 3 + m) * KCH + k];
  }
  for (int idx = tid; idx < NPB * 5 * KCH; idx += NTHREADS) {
    int k = idx % KCH; int m = (idx / KCH) % 5; int node = idx / (5 * KCH);
    feat[(node * 9 + 4 + m) * KCH + k] = f2[((size_t)(node0 + node) * 5 + m) * KCH + k];
  }
  __syncthreads();

  for (int it = 0; it < NITER; ++it) {
    // ---- Stage 1: equivariant RMS norm over m + gamma gain -> an (m-major rows)
    for (int idx = tid; idx < NPB * 3 * KCH; idx += NTHREADS) {
      int k = idx % KCH; int l = (idx / KCH) % 3; int node = idx / (3 * KCH);
      int mb = (l == 0) ? 0 : ((l == 1) ? 1 : 4);
      int mc = 2 * l + 1;
      float ss = 0.f;
      for (int mm = 0; mm < mc; ++mm) {
        float v = feat[(node * 9 + mb + mm) * KCH + k];
        ss = fmaf(v, v, ss);
      }
      float sc = rsqrtf(ss / (float)mc + EPSV) * gamma[((size_t)it * 3 + l) * KCH + k];
      for (int mm = 0; mm < mc; ++mm) {
        int m = mb + mm;
        an[(m * NPB + node) * KCH + k] = feat[(node * 9 + m) * KCH + k] * sc;
      }
    }
    __syncthreads();

    // ---- Stage 2: linear_in  an[144x128] x Win[it][l][128x256] -> hc[144x256]
    //      two independent accumulator chains for WMMA ILP
    for (int tile = wave; tile < 9 * 16; tile += NWAVES) {
      int mt = tile >> 4;            // m-tile == m (9)
      int nt = tile & 15;            // p-tile (16)
      int l  = l_of_m(mt);
      const float* W    = Win + ((size_t)it * 3 + l) * KCH * PCH + nt * 16 + row16;
      const float* arow = an + (mt * NPB + row16) * KCH + khalf;
      v8f acc0 = {0.f,0.f,0.f,0.f,0.f,0.f,0.f,0.f};
      v8f acc1 = {0.f,0.f,0.f,0.f,0.f,0.f,0.f,0.f};
      #pragma unroll 4
      for (int kk = 0; kk < KCH; kk += 8) {
        v2f a0, b0, a1, b1;
        a0.x = arow[kk];     a0.y = arow[kk + 1];
        a1.x = arow[kk + 4]; a1.y = arow[kk + 5];
        b0.x = W[(size_t)(kk + khalf)     * PCH];
        b0.y = W[(size_t)(kk + khalf + 1) * PCH];
        b1.x = W[(size_t)(kk + 4 + khalf)     * PCH];
        b1.y = W[(size_t)(kk + 4 + khalf + 1) * PCH];
        acc0 = __builtin_amdgcn_wmma_f32_16x16x4_f32(false, a0, false, b0,
                                                     (short)0, acc0, false, false);
        acc1 = __builtin_amdgcn_wmma_f32_16x16x4_f32(false, a1, false, b1,
                                                     (short)0, acc1, false, false);
      }
      #pragma unroll
      for (int j = 0; j < 8; ++j)
        hc[(mt * NPB + mibase + j) * PCH + nt * 16 + row16] = acc0[j] + acc1[j];
    }
    __syncthreads();

    // ---- Stage 3: CG tensor product as WMMA GEMM.
    //      Rows = (node,p) columns of hc; K = 45 symmetric pairs (pad 48); N = 16 (c<9 live).
    //      Per-lane VALU: 45 products + selects (vs 729 FMA scalar).
    {
      // Preload the 12 shared B fragments of Ub (invariant across tiles and iterations)
      v2f bU[12];
      #pragma unroll
      for (int s = 0; s < 12; ++s) {
        bU[s].x = Ub[(4 * s + khalf)     * 16 + row16];
        bU[s].y = Ub[(4 * s + khalf + 1) * 16 + row16];
      }
      for (int tile = wave; tile < NPB * 16; tile += NWAVES) {
        int node  = tile >> 4;
        int pbase = (tile & 15) * 16;
        // each half-wave lane owns pair-row row16 -> (node, p = pbase+row16)
        float v[9];
        #pragma unroll
        for (int a = 0; a < 9; ++a)
          v[a] = hc[(a * NPB + node) * PCH + pbase + row16];
        float prods[48];
        #pragma unroll
        for (int q = 0; q < 45; ++q) prods[q] = v[PA[q]] * v[PB[q]];
        prods[45] = prods[46] = prods[47] = 0.f;
        v8f acc0 = {0.f,0.f,0.f,0.f,0.f,0.f,0.f,0.f};
        v8f acc1 = {0.f,0.f,0.f,0.f,0.f,0.f,0.f,0.f};
        #pragma unroll
        for (int s = 0; s < 12; ++s) {
          int kk = 4 * s;
          v2f a;
          a.x = hiHalf ? prods[kk + 2] : prods[kk + 0];
          a.y = hiHalf ? prods[kk + 3] : prods[kk + 1];
          if (s & 1)
            acc1 = __builtin_amdgcn_wmma_f32_16x16x4_f32(false, a, false, bU[s],
                                                         (short)0, acc1, false, false);
          else
            acc0 = __builtin_amdgcn_wmma_f32_16x16x4_f32(false, a, false, bU[s],
                                                         (short)0, acc0, false, false);
        }
        // D[Mi][Ni]: Mi = pair row (p = pbase+Mi), Ni = c.  In-place column update.
        if (row16 < 9) {
          #pragma unroll
          for (int j = 0; j < 8; ++j)
            hc[(row16 * NPB + node) * PCH + pbase + mibase + j] = acc0[j] + acc1[j];
        }
      }
    }
    __syncthreads();

    // ---- Stage 4: linear_out  hc[144x256] x Wout[it][l][256x128] -> feat += (residual)
    for (int tile = wave; tile < 9 * 8; tile += NWAVES) {
      int mt = tile >> 3;            // m-tile (9)
      int nt = tile & 7;             // k-tile (8)
      int l  = l_of_m(mt);
      const float* W    = Wout + ((size_t)it * 3 + l) * PCH * KCH + nt * 16 + row16;
      const float* arow = hc + (mt * NPB + row16) * PCH + khalf;
      v8f acc0 = {0.f,0.f,0.f,0.f,0.f,0.f,0.f,0.f};
      v8f acc1 = {0.f,0.f,0.f,0.f,0.f,0.f,0.f,0.f};
      #pragma unroll 4
      for (int kk = 0; kk < PCH; kk += 8) {
        v2f a0, b0, a1, b1;
        a0.x = arow[kk];     a0.y = arow[kk + 1];
        a1.x = arow[kk + 4]; a1.y = arow[kk + 5];
        b0.x = W[(size_t)(kk + khalf)     * KCH];
        b0.y = W[(size_t)(kk + khalf + 1) * KCH];
        b1.x = W[(size_t)(kk + 4 + khalf)     * KCH];
        b1.y = W[(size_t)(kk + 4 + khalf + 1) * KCH];
        acc0 = __builtin_amdgcn_wmma_f32_16x16x4_f32(false, a0, false, b0,
                                                     (short)0, acc0, false, false);
        acc1 = __builtin_amdgcn_wmma_f32_16x16x4_f32(false, a1, false, b1,
                                                     (short)0, acc1, false, false);
      }
      #pragma unroll
      for (int j = 0; j < 8; ++j) {
        int node = mibase + j;
        feat[(node * 9 + mt) * KCH + nt * 16 + row16] += acc0[j] + acc1[j];
      }
    }
    __syncthreads();
  }

  // ---- Write output: [N, 9, K] packed over l == our node-major feat layout
  float* outb = out + (size_t)node0 * 9 * KCH;
  for (int idx = tid; idx < NPB * 9 * KCH; idx += NTHREADS)
    outb[idx] = feat[idx];
}

extern "C" void kernel_launch(void* const* d_in, const int* in_sizes, int n_in,
                              void* d_out, int out_size, void* d_ws, size_t ws_size,
                              hipStream_t stream) {
  (void)in_sizes; (void)n_in; (void)out_size; (void)d_ws; (void)ws_size;
  const float* f0 = (const float*)d_in[0];
  const float* f1 = (const float*)d_in[1];
  const float* f2 = (const float*)d_in[2];
  const float* U  = (const float*)d_in[3];
  const float* g  = (const float*)d_in[4];
  const float* Wi = (const float*)d_in[5];
  const float* Wo = (const float*)d_in[6];
  float* out = (float*)d_out;

  size_t shmem = (size_t)(NPB * 9 * KCH   // feat      (72 KB)
                        + 9 * NPB * KCH   // an        (72 KB)
                        + 9 * NPB * PCH   // hc        (144 KB)
                        + 48 * 16) * sizeof(float);   // Ub (3 KB)  => ~292 KB < 320 KB WGP LDS

  dim3 grid(NN / NPB);   // 3125 blocks of 16 nodes
  dim3 block(NTHREADS);
  cg_fused_kernel<<<grid, block, shmem, stream>>>(f0, f1, f2, U, g, Wi, Wo, out);
}